// MagnitudeWarp_61864708931684
// MI455X (gfx1250) — compile-verified
//
#include <hip/hip_runtime.h>

// MagnitudeWarp for MI455X (gfx1250, wave32).
//
// out[b,s,f] = x[b,s,f] * warp[b,s,f],  warp = cubic Lagrange interp of
// kv[b,f,0..3] at knots u in {0,1/3,2/3,1}, evaluated at u = s/(S-1).
//
// Bandwidth-bound (~256 MB @ 23.3 TB/s ~= 11 us). The warp construction is a
// K=4 f32 GEMM which maps exactly onto V_WMMA_F32_16X16X4_F32; the WMMA C/D
// lane layout is used directly as the addressing pattern for coalesced
// x-loads / out-stores (dword per lane, 64B per half-wave row).

typedef float v2f __attribute__((ext_vector_type(2)));
typedef float v8f __attribute__((ext_vector_type(8)));

namespace {
constexpr int kB = 64;
constexpr int kS = 4096;
constexpr int kF = 128;
constexpr int kK = 4;
constexpr int kSTilesPerBlock = 16;   // 256 s-rows per block
}

__global__ __launch_bounds__(256) void MagnitudeWarp_wmma_kernel(
    const float* __restrict__ x,       // [B, S, F]
    const float* __restrict__ kv,      // [B, F, K]
    float* __restrict__ out) {         // [B, S, F]
  const int tid  = threadIdx.x;
  const int wave = tid >> 5;           // 0..7 -> feature tile
  const int lane = tid & 31;
  const int half = lane >> 4;          // 0: K=0,1 pair; 1: K=2,3 pair
  const int l16  = lane & 15;          // M row (A) / N col (B, C/D)

  const int b       = blockIdx.x;                          // batch
  const int s_block = blockIdx.y * (16 * kSTilesPerBlock); // base s row
  const int f0      = wave * 16;                           // feature tile base

  // ---- B matrix: kv^T (4x16 f32, 2 VGPRs). Lane l16 = column N = feature
  // f0+l16. VGPR0/1 hold K=0/1 (lanes 0-15) and K=2/3 (lanes 16-31).
  const float* kvp = kv + (((size_t)b * kF + (f0 + l16)) * kK + half * 2);
  v2f Bm;
  Bm.x = kvp[0];
  Bm.y = kvp[1];

  // Per-lane x/out base address for the C/D layout:
  //   VGPR v of D holds element (M = v + 8*half, N = l16) of the tile.
  const size_t xbase =
      ((size_t)b * kS + (size_t)(s_block + half * 8)) * kF + (f0 + l16);

  const float inv = 1.0f / (float)(kS - 1);

  for (int t = 0; t < kSTilesPerBlock; ++t) {
    const int s0 = s_block + t * 16;

    // ---- A matrix: basis tile (16x4 f32, 2 VGPRs). Row M = l16 at
    // u = (s0+M)/(S-1). Lanes 0-15 supply Lagrange L0/L1, lanes 16-31 L2/L3.
    const float u    = (float)(s0 + l16) * inv;
    const float um1  = u - 1.0f;
    const float um13 = u - (1.0f / 3.0f);
    const float um23 = u - (2.0f / 3.0f);
    const float L0 = -4.5f  * um13 * um23 * um1;
    const float L1 =  13.5f * u    * um23 * um1;
    const float L2 = -13.5f * u    * um13 * um1;
    const float L3 =  4.5f  * u    * um13 * um23;
    v2f Am;
    Am.x = half ? L2 : L0;
    Am.y = half ? L3 : L1;

    // warp tile (16x16 f32) = A(16x4) x B(4x16); C = 0.
    v8f c = {};
    c = __builtin_amdgcn_wmma_f32_16x16x4_f32(
        /*neg_a=*/false, Am, /*neg_b=*/false, Bm,
        /*c_mod=*/(short)0, c, /*reuse_a=*/false, /*reuse_b=*/false);

    // Elementwise multiply against x and store, reusing the D layout as the
    // address pattern (coalesced dword-per-lane across each 16-lane half).
    const size_t base = xbase + (size_t)t * 16 * kF;
#pragma unroll
    for (int v = 0; v < 8; ++v) {
      const size_t idx = base + (size_t)v * kF;
      out[idx] = x[idx] * c[v];
    }
  }
}

extern "C" void kernel_launch(void* const* d_in, const int* in_sizes, int n_in,
                              void* d_out, int out_size, void* d_ws,
                              size_t ws_size, hipStream_t stream) {
  (void)in_sizes; (void)n_in; (void)out_size; (void)d_ws; (void)ws_size;
  const float* x  = (const float*)d_in[0];   // [64, 4096, 128] f32
  const float* kv = (const float*)d_in[1];   // [64, 128, 4]   f32
  float* out      = (float*)d_out;           // [64, 4096, 128] f32

  dim3 grid(kB, kS / (16 * kSTilesPerBlock));  // (64, 16)
  dim3 block(256);                             // 8 wave32 waves
  MagnitudeWarp_wmma_kernel<<<grid, block, 0, stream>>>(x, kv, out);
}